// VirtualNodeGNNOriginal_52879637348572
// MI455X (gfx1250) — compile-verified
//
#include <hip/hip_runtime.h>
#include <hip/hip_bf16.h>

typedef float v8f __attribute__((ext_vector_type(8)));
typedef float v2f __attribute__((ext_vector_type(2)));

#define FEAT 128

// ---------------------------------------------------------------------------
// Degree / normalization kernels
// ---------------------------------------------------------------------------
__global__ void deg_init_kernel(float* __restrict__ deg, int n) {
    long i = (long)blockIdx.x * blockDim.x + threadIdx.x;
    if (i < n) deg[i] = 1.0f;  // self-loop contributes 1 to every node's degree
}

__global__ void deg_accum_kernel(const int* __restrict__ dst, float* __restrict__ deg, int nE) {
    long e = (long)blockIdx.x * blockDim.x + threadIdx.x;
    if (e < nE) unsafeAtomicAdd(&deg[dst[e]], 1.0f);
}

__global__ void dinv_kernel(float* __restrict__ deg_dinv, int n) {
    long i = (long)blockIdx.x * blockDim.x + threadIdx.x;
    if (i < n) {
        float d = deg_dinv[i];
        deg_dinv[i] = (d > 0.0f) ? rsqrtf(d) : 0.0f;  // in-place deg -> deg^-1/2
    }
}

// ---------------------------------------------------------------------------
// GEMM: H[nrows x 128] = X[nrows x 128] @ W[128 x 128]
// fp32 WMMA 16x16x4. One wave owns a 16-row tile, 8 column tiles of 16.
// W staged in LDS (64 KB). Row-tile guard is wave-uniform -> EXEC all ones.
// ---------------------------------------------------------------------------
__global__ void gcn_gemm128_kernel(const float* __restrict__ X,
                                   const float* __restrict__ W,
                                   float* __restrict__ H, int nrows) {
    __shared__ float Ws[FEAT * FEAT];  // 64 KB

    const int tid = threadIdx.x;  // 256 threads = 8 waves
    // cooperative load of W into LDS as float4
    {
        const float4* Wv = (const float4*)W;
        float4* Wsv = (float4*)Ws;
        #pragma unroll
        for (int i = 0; i < (FEAT * FEAT / 4) / 256; ++i)
            Wsv[tid + i * 256] = Wv[tid + i * 256];
    }
    __syncthreads();

    const int wave = tid >> 5;
    const int lane = tid & 31;
    const int half = lane >> 4;     // 0: K pair {0,1}, 1: K pair {2,3}
    const int lm   = lane & 15;     // M (for A) / N (for B) index

    const long rowTile = (long)blockIdx.x * 8 + wave;
    const long row0 = rowTile * 16;
    if (row0 >= nrows) return;      // wave-uniform

    v8f acc[8] = {};

    const float* Xrow = X + (row0 + lm) * FEAT;

    for (int k = 0; k < FEAT; k += 4) {
        // A fragment: 16x4 f32 -> 2 VGPRs/lane.
        // lanes 0-15: K = k,k+1 ; lanes 16-31: K = k+2,k+3  (ISA layout)
        const float2 av = *(const float2*)(Xrow + k + 2 * half);
        v2f a; a.x = av.x; a.y = av.y;

        #pragma unroll
        for (int t = 0; t < 8; ++t) {
            // B fragment: 4x16 f32, mirrored layout (N across lanes, K pairs in VGPRs)
            v2f b;
            b.x = Ws[(k + 2 * half) * FEAT + t * 16 + lm];
            b.y = Ws[(k + 2 * half + 1) * FEAT + t * 16 + lm];
            acc[t] = __builtin_amdgcn_wmma_f32_16x16x4_f32(
                /*neg_a=*/false, a, /*neg_b=*/false, b,
                /*c_mod=*/(short)0, acc[t], /*reuse_a=*/false, /*reuse_b=*/false);
        }
    }

    // C/D layout: VGPR r -> M = r + 8*half, N = lm
    #pragma unroll
    for (int t = 0; t < 8; ++t) {
        #pragma unroll
        for (int r = 0; r < 8; ++r) {
            const int m = r + 8 * half;
            H[(row0 + m) * FEAT + t * 16 + lm] = acc[t][r];
        }
    }
}

// ---------------------------------------------------------------------------
// agg init: AGG[i,f] = b[f] + dinv[i]^2 * H[i,f]   (bias + self-loop term)
// ---------------------------------------------------------------------------
__global__ void agg_init_kernel(const float* __restrict__ H,
                                const float* __restrict__ dinv,
                                const float* __restrict__ bias,
                                float* __restrict__ AGG, long total) {
    long gid = (long)blockIdx.x * blockDim.x + threadIdx.x;
    if (gid >= total) return;
    long node = gid >> 7;
    int f = (int)(gid & (FEAT - 1));
    float w = dinv[node];
    AGG[gid] = bias[f] + w * w * H[gid];
}

// ---------------------------------------------------------------------------
// Edge scatter: one wave per edge, each lane handles 4 features (float4 gather,
// 4 hardware f32 atomics into L2-resident AGG).
// ---------------------------------------------------------------------------
__global__ void scatter_kernel(const int* __restrict__ src,
                               const int* __restrict__ dst,
                               const float* __restrict__ dinv,
                               const float* __restrict__ H,
                               float* __restrict__ AGG, int nE) {
    long gid = (long)blockIdx.x * blockDim.x + threadIdx.x;
    long e = gid >> 5;
    if (e >= nE) return;
    const int lane = (int)(gid & 31);
    const int s = src[e];
    const int d = dst[e];
    const float w = dinv[s] * dinv[d];
    const float4 hv = ((const float4*)(H + (long)s * FEAT))[lane];
    float* out = AGG + (long)d * FEAT + lane * 4;
    unsafeAtomicAdd(out + 0, hv.x * w);
    unsafeAtomicAdd(out + 1, hv.y * w);
    unsafeAtomicAdd(out + 2, hv.z * w);
    unsafeAtomicAdd(out + 3, hv.w * w);
}

// ---------------------------------------------------------------------------
// ReLU: X[i] = max(AGG[i], 0)
// ---------------------------------------------------------------------------
__global__ void relu_kernel(const float* __restrict__ AGG, float* __restrict__ X, long total) {
    long gid = (long)blockIdx.x * blockDim.x + threadIdx.x;
    if (gid < total) X[gid] = fmaxf(AGG[gid], 0.0f);
}

// ---------------------------------------------------------------------------
// Final FC on last node: out[j] = x[N-1] . fcW[:,j] + fcb[j], j in [0,64)
// ---------------------------------------------------------------------------
__global__ void fc_out_kernel(const float* __restrict__ X,
                              const float* __restrict__ fcW,
                              const float* __restrict__ fcb,
                              float* __restrict__ out, int nrows, int dout) {
    int j = threadIdx.x;
    if (j >= dout) return;
    const float* xr = X + (long)(nrows - 1) * FEAT;
    float acc = fcb[j];
    #pragma unroll 8
    for (int k = 0; k < FEAT; ++k) acc = fmaf(xr[k], fcW[k * dout + j], acc);
    out[j] = acc;
}

// ---------------------------------------------------------------------------
// Host-side launcher (graph-capture safe: only kernel launches on `stream`)
// ---------------------------------------------------------------------------
extern "C" void kernel_launch(void* const* d_in, const int* in_sizes, int n_in,
                              void* d_out, int out_size, void* d_ws, size_t ws_size,
                              hipStream_t stream) {
    const float* x   = (const float*)d_in[0];
    const int*   ei  = (const int*)d_in[1];
    const float* W1  = (const float*)d_in[2];
    const float* b1  = (const float*)d_in[3];
    const float* W2  = (const float*)d_in[4];
    const float* b2  = (const float*)d_in[5];
    const float* W3  = (const float*)d_in[6];
    const float* b3  = (const float*)d_in[7];
    const float* fcW = (const float*)d_in[8];
    const float* fcb = (const float*)d_in[9];

    const int N = in_sizes[0] / FEAT;
    const int E = in_sizes[1] / 2;
    const int OUTD = out_size;

    const int* srcp = ei;
    const int* dstp = ei + E;

    // Workspace layout: dinv[N] | X[N*128] | H[N*128] | AGG[N*128]
    char* ws = (char*)d_ws;
    size_t off = 0;
    float* dinv = (float*)(ws + off); off += (((size_t)N * 4) + 255) & ~(size_t)255;
    float* Xb   = (float*)(ws + off); off += (size_t)N * FEAT * 4;
    float* Hb   = (float*)(ws + off); off += (size_t)N * FEAT * 4;
    float* AGG  = (float*)(ws + off);

    const long total = (long)N * FEAT;

    // --- normalization ---
    deg_init_kernel<<<(N + 255) / 256, 256, 0, stream>>>(dinv, N);
    deg_accum_kernel<<<(E + 255) / 256, 256, 0, stream>>>(dstp, dinv, E);
    dinv_kernel<<<(N + 255) / 256, 256, 0, stream>>>(dinv, N);

    const float* Wl[3] = {W1, W2, W3};
    const float* bl[3] = {b1, b2, b3};

    const int rowTiles  = (N + 15) / 16;
    const int gemmBlks  = (rowTiles + 7) / 8;
    const long scatThreads = (long)E * 32;
    const int scatBlks  = (int)((scatThreads + 255) / 256);
    const int elemBlks  = (int)((total + 255) / 256);

    const float* xin = x;
    for (int l = 0; l < 3; ++l) {
        gcn_gemm128_kernel<<<gemmBlks, 256, 0, stream>>>(xin, Wl[l], Hb, N);
        agg_init_kernel<<<elemBlks, 256, 0, stream>>>(Hb, dinv, bl[l], AGG, total);
        scatter_kernel<<<scatBlks, 256, 0, stream>>>(srcp, dstp, dinv, Hb, AGG, E);
        relu_kernel<<<elemBlks, 256, 0, stream>>>(AGG, Xb, total);
        xin = Xb;
    }

    fc_out_kernel<<<1, 64, 0, stream>>>(Xb, fcW, fcb, (float*)d_out, N, OUTD);
}